// Attn_78640851190325
// MI455X (gfx1250) — compile-verified
//
#include <hip/hip_runtime.h>

// ---------------------------------------------------------------------------
// Attention scores + softmax for MI455X (gfx1250, wave32).
//
// scores = E(65536x1024, f32) . h(1024, f32)  -> softmax over 65536
//
// Roofline: streams 256MB of E once from HBM -> ~11us floor at 23.3 TB/s.
// Bandwidth-bound; fp32 kept end-to-end via V_WMMA_F32_16X16X4_F32:
// one wave owns 16 rows; A-fragment = 16x4 fp32 tile (ISA layout: lane=M,
// {VGPR,half-wave}=K), B-fragment = the 4-entry hidden chunk replicated over
// all 16 N columns (lane-uniform per half-wave, same K mapping as A), so every
// column of D equals the row dot product; the v8f accumulator carries it
// across 1024/4 K-steps.
//
// Latency plan: unroll 16 (16 outstanding b64 A-loads/wave ~= 4KB wave-wide in
// flight) + global_prefetch_b8 one 64B line per row at +1KB, 4x per unrolled
// body (matches the 256B/row consumed per body). hidden lives in LDS so
// B-fragments ride DScnt instead of cluttering the A-stream LOADcnt chain.
// Softmax uses a parallel online (max,sum) reduction: deterministic tree.
// ---------------------------------------------------------------------------

typedef __attribute__((ext_vector_type(2))) float v2f;
typedef __attribute__((ext_vector_type(4))) float v4f;
typedef __attribute__((ext_vector_type(8))) float v8f;

#define SEQ_LEN 65536
#define HIDDEN  1024
#define RBLOCKS 64   // partial-reduce blocks (1024 scores each)

__global__ __launch_bounds__(256) void attn_scores_wmma_f32(
    const float* __restrict__ enc,     // [SEQ_LEN, HIDDEN]
    const float* __restrict__ hidden,  // [HIDDEN]
    float* __restrict__ scores)        // [SEQ_LEN]
{
  __shared__ float sh[HIDDEN];         // 4KB: block-shared copy of hidden

  // Cooperative fill: 256 threads x one b128 each.
  ((v4f*)sh)[threadIdx.x] = ((const v4f*)hidden)[threadIdx.x];
  __syncthreads();

  const int lane = threadIdx.x & 31;
  const int wave = (blockIdx.x * (blockDim.x >> 5)) + (threadIdx.x >> 5);
  const int tile = wave * 16;                // 16 rows per wave
  const int m    = lane & 15;                // row within tile
  const int koff = (lane >> 4) << 1;         // 0 (lanes 0-15) or 2 (lanes 16-31)

  const float* aPtr = enc + (size_t)(tile + m) * HIDDEN + koff;
  const float* bPtr = sh + koff;

  v8f acc = {};
#pragma unroll 16
  for (int k = 0; k < HIDDEN; k += 4) {
    if ((k & 15) == 0) {
      // one 64B line per row, 1KB ahead; 4 statically-placed per unrolled
      // body, advancing 64B each -> exactly covers this body's consumption.
      __builtin_prefetch(aPtr + k + 256, 0, 3);   // global_prefetch_b8
    }
    v2f a = *(const v2f*)(aPtr + k);   // A frag: lane's 2 K-values of its row
    v2f b = *(const v2f*)(bPtr + k);   // B frag from LDS (ds_load_b64)
    acc = __builtin_amdgcn_wmma_f32_16x16x4_f32(
        /*neg_a=*/false, a, /*neg_b=*/false, b,
        /*c_mod=*/(short)0, acc, /*reuse_a=*/false, /*reuse_b=*/false);
  }

  // D layout (32-bit 16x16): acc[r] = row (tile+r) on lanes 0-15,
  //                          acc[r] = row (tile+8+r) on lanes 16-31.
  if (lane == 0) {
    v4f* o = (v4f*)(scores + tile);
    o[0] = (v4f){acc[0], acc[1], acc[2], acc[3]};
    o[1] = (v4f){acc[4], acc[5], acc[6], acc[7]};
  } else if (lane == 16) {
    v4f* o = (v4f*)(scores + tile + 8);
    o[0] = (v4f){acc[0], acc[1], acc[2], acc[3]};
    o[1] = (v4f){acc[4], acc[5], acc[6], acc[7]};
  }
}

// Online-softmax combine: (m,s) <- merge of two partial (max, sum-exp) pairs.
__device__ __forceinline__ void osm_combine(float& m0, float& s0,
                                            float m1, float s1) {
  float mm = fmaxf(m0, m1);
  s0 = s0 * __expf(m0 - mm) + s1 * __expf(m1 - mm);
  m0 = mm;
}

// Phase 2a: 64 blocks, each produces one (max, sum-exp) pair over 1024 scores.
__global__ __launch_bounds__(256) void attn_partial_reduce(
    const float* __restrict__ scores, float* __restrict__ partials)
{
  __shared__ float sm[256];
  __shared__ float ss[256];
  const int tid  = threadIdx.x;
  const int base = blockIdx.x * 1024;

  float v0 = scores[base + tid];
  float v1 = scores[base + 256 + tid];
  float v2 = scores[base + 512 + tid];
  float v3 = scores[base + 768 + tid];
  float m = fmaxf(fmaxf(v0, v1), fmaxf(v2, v3));
  float s = __expf(v0 - m) + __expf(v1 - m) + __expf(v2 - m) + __expf(v3 - m);
  sm[tid] = m; ss[tid] = s;
  __syncthreads();
  for (int st = 128; st > 0; st >>= 1) {
    if (tid < st) {
      float mm = sm[tid], sc = ss[tid];
      osm_combine(mm, sc, sm[tid + st], ss[tid + st]);
      sm[tid] = mm; ss[tid] = sc;
    }
    __syncthreads();
  }
  if (tid == 0) {
    partials[2 * blockIdx.x]     = sm[0];
    partials[2 * blockIdx.x + 1] = ss[0];
  }
}

// Phase 2b: fold the 64 pairs -> params = {gmax, 1/sum}. Deterministic tree.
__global__ __launch_bounds__(RBLOCKS) void attn_combine(
    const float* __restrict__ partials, float* __restrict__ params)
{
  __shared__ float sm[RBLOCKS];
  __shared__ float ss[RBLOCKS];
  const int tid = threadIdx.x;
  sm[tid] = partials[2 * tid];
  ss[tid] = partials[2 * tid + 1];
  __syncthreads();
  for (int st = RBLOCKS / 2; st > 0; st >>= 1) {
    if (tid < st) {
      float mm = sm[tid], sc = ss[tid];
      osm_combine(mm, sc, sm[tid + st], ss[tid + st]);
      sm[tid] = mm; ss[tid] = sc;
    }
    __syncthreads();
  }
  if (tid == 0) {
    params[0] = sm[0];
    params[1] = 1.0f / ss[0];
  }
}

// Phase 3: in-place normalize: out[i] = exp(out[i] - max) / sum
__global__ __launch_bounds__(256) void attn_softmax_norm(
    float* __restrict__ out, const float* __restrict__ params)
{
  const int i = blockIdx.x * blockDim.x + threadIdx.x;
  const float gmax = params[0];
  const float rsum = params[1];
  out[i] = __expf(out[i] - gmax) * rsum;
}

extern "C" void kernel_launch(void* const* d_in, const int* in_sizes, int n_in,
                              void* d_out, int out_size, void* d_ws, size_t ws_size,
                              hipStream_t stream) {
  const float* hidden = (const float*)d_in[0];  // [1024]
  const float* enc    = (const float*)d_in[1];  // [65536, 1024]
  float* out          = (float*)d_out;          // [65536] (= [1,1,65536])

  float* partials = (float*)d_ws;               // 64 x (max, sum)  = 512B
  float* params   = (float*)d_ws + 2 * RBLOCKS; // {gmax, 1/sum}

  // Phase 1: scores via fp32 WMMA. 65536/16 = 4096 waves = 512 blocks x 8 waves.
  attn_scores_wmma_f32<<<512, 256, 0, stream>>>(enc, hidden, out);

  // Phase 2: parallel online (max, sum-exp) reduction over L2-hot scores.
  attn_partial_reduce<<<RBLOCKS, 256, 0, stream>>>(out, partials);
  attn_combine<<<1, RBLOCKS, 0, stream>>>(partials, params);

  // Phase 3: normalize in place. 65536 / 256 = 256 blocks.
  attn_softmax_norm<<<SEQ_LEN / 256, 256, 0, stream>>>(out, params);
}